// ReprojNonBatched_89386859364357
// MI455X (gfx1250) — compile-verified
//
#include <hip/hip_runtime.h>
#include <stdint.h>

// Bundle-adjustment reprojection residuals on MI455X (gfx1250).
// Bandwidth-bound gather/stream kernel: camera table staged in LDS via the
// CDNA5 async global->LDS path (inline asm: global_load_async_to_lds_b32 +
// s_wait_asynccnt); hot per-observation camera reads are wide DS loads;
// streaming traffic uses non-temporal cache hints.

#define BLOCK      256
#define MAX_CAMS   1024
#define CAM_STRIDE 12   // 10 payload floats padded to 48B for b128-aligned DS reads

typedef float v2f __attribute__((ext_vector_type(2)));
typedef float v4f __attribute__((ext_vector_type(4)));

#if defined(__gfx1250__)
#define USE_ASYNC_LDS 1
#else
#define USE_ASYNC_LDS 0
#endif

#if USE_ASYNC_LDS
// Async copy 4 bytes: from global address gptr + OFF to LDS address lds_off + OFF; tracked by ASYNCcnt.
// IOFFSET is applied to both global and LDS addresses (ISA 15.18.3 / CDNA5 ISA).
#define ASYNC_B32(lds_off, gptr, OFF)                                          \
  asm volatile("global_load_async_to_lds_b32 %0, %1, off offset:" #OFF         \
               :: "v"(lds_off), "v"(gptr) : "memory")
#endif

__global__ __launch_bounds__(BLOCK) void ReprojResidual_kernel(
    const float* __restrict__ camp,   // [n_cams,7]  qx qy qz qw tx ty tz
    const float* __restrict__ pts,    // [n_pts,3]
    const float* __restrict__ obs,    // [n_obs,2]
    const float* __restrict__ intr,   // [n_cams,3]  f k1 k2
    const int*   __restrict__ cidx,   // [n_obs]
    const int*   __restrict__ pidx,   // [n_obs]
    float*       __restrict__ out,    // [n_obs,2]
    int n_cams, int n_obs)
{
  // LDS camera table: row = {qx,qy,qz,qw, tx,ty,tz, f, k1,k2, pad,pad}
  __shared__ __align__(16) float s_cam[MAX_CAMS * CAM_STRIDE];

  // ---- Stage camera params + intrinsics into LDS (one-time per block) ----
#if USE_ASYNC_LDS
  for (int c = threadIdx.x; c < n_cams; c += BLOCK) {
    const float* cp = camp + 7 * c;
    const float* ip = intr + 3 * c;
    float* dst = &s_cam[c * CAM_STRIDE];
    // Generic LDS address: low 32 bits are the LDS byte offset (aperture map).
    uint32_t lc = (uint32_t)(uintptr_t)dst;        // row base
    uint32_t li = lc + 28;                         // row base + 7 floats
    ASYNC_B32(lc, cp, 0);
    ASYNC_B32(lc, cp, 4);
    ASYNC_B32(lc, cp, 8);
    ASYNC_B32(lc, cp, 12);
    ASYNC_B32(lc, cp, 16);
    ASYNC_B32(lc, cp, 20);
    ASYNC_B32(lc, cp, 24);
    ASYNC_B32(li, ip, 0);
    ASYNC_B32(li, ip, 4);
    ASYNC_B32(li, ip, 8);
  }
  asm volatile("s_wait_asynccnt 0" ::: "memory");
  __syncthreads();
#else
  for (int c = threadIdx.x; c < n_cams; c += BLOCK) {
    const float* cp = camp + 7 * c;
    const float* ip = intr + 3 * c;
    float* dst = &s_cam[c * CAM_STRIDE];
#pragma unroll
    for (int j = 0; j < 7; ++j) dst[j] = cp[j];
    dst[7] = ip[0]; dst[8] = ip[1]; dst[9] = ip[2];
  }
  __syncthreads();
#endif

  const v2f* __restrict__ obs2 = reinterpret_cast<const v2f*>(obs);
  v2f* __restrict__ out2 = reinterpret_cast<v2f*>(out);

  const int stride = gridDim.x * BLOCK;
  for (int i = blockIdx.x * BLOCK + threadIdx.x; i < n_obs; i += stride) {
    // Streaming (read-once) data: non-temporal loads.
    const int c = __builtin_nontemporal_load(cidx + i);
    const int p = __builtin_nontemporal_load(pidx + i);
    const v2f o = __builtin_nontemporal_load(obs2 + i);

    // Camera record from LDS: b128 + b128 + b64 DS loads (rows 16B aligned).
    const float* row = &s_cam[c * CAM_STRIDE];
    const v4f q  = *reinterpret_cast<const v4f*>(row);      // qx qy qz qw
    const v4f tf = *reinterpret_cast<const v4f*>(row + 4);  // tx ty tz f
    const v2f kk = *reinterpret_cast<const v2f*>(row + 8);  // k1 k2

    // Point gather (3 MB point cloud is L2-resident).
    const float* pp = pts + 3 * p;
    const float px = pp[0], py = pp[1], pz = pp[2];

    // a = p - t
    const float ax = px - tf.x, ay = py - tf.y, az = pz - tf.z;
    // Rotate by conjugate quaternion: qv = -q.xyz, w = q.w
    const float vx = -q.x, vy = -q.y, vz = -q.z;
    // uv = cross(qv, a)
    const float ux = vy * az - vz * ay;
    const float uy = vz * ax - vx * az;
    const float uz = vx * ay - vy * ax;
    // w2 = cross(qv, uv)
    const float wx = vy * uz - vz * uy;
    const float wy = vz * ux - vx * uz;
    const float wz = vx * uy - vy * ux;
    const float rx = ax + 2.0f * (q.w * ux + wx);
    const float ry = ay + 2.0f * (q.w * uy + wy);
    const float rz = az + 2.0f * (q.w * uz + wz);

    const float invz = 1.0f / rz;           // exact IEEE divide (validation-safe)
    const float x = rx * invz;
    const float y = ry * invz;

    const float n = x * x + y * y;
    const float r = 1.0f + kk.x * n + kk.y * n * n;
    const float s = r * tf.w;               // r * f

    v2f res;
    res.x = x * s - o.x;
    res.y = y * s - o.y;
    __builtin_nontemporal_store(res, out2 + i);  // write-once output: NT store
  }
}

extern "C" void kernel_launch(void* const* d_in, const int* in_sizes, int n_in,
                              void* d_out, int out_size, void* d_ws, size_t ws_size,
                              hipStream_t stream) {
  const float* camp = (const float*)d_in[0];   // camera_params [n_cams,7]
  const float* pts  = (const float*)d_in[1];   // points_3d    [n_pts,3]
  const float* obs  = (const float*)d_in[2];   // points_2d    [n_obs,2]
  const float* intr = (const float*)d_in[3];   // intr         [n_cams,3]
  const int*   cidx = (const int*)d_in[4];     // camera_indices [n_obs]
  const int*   pidx = (const int*)d_in[5];     // point_indices  [n_obs]
  float* out = (float*)d_out;

  const int n_cams = in_sizes[0] / 7;
  const int n_obs  = in_sizes[4];

  // Grid-stride: enough blocks to saturate the chip, few enough that the
  // per-block 40KB LDS staging stays negligible vs. the 96MB HBM stream.
  int blocks = (n_obs + BLOCK - 1) / BLOCK;
  if (blocks > 2048) blocks = 2048;
  if (blocks < 1) blocks = 1;

  hipLaunchKernelGGL(ReprojResidual_kernel, dim3(blocks), dim3(BLOCK), 0, stream,
                     camp, pts, obs, intr, cidx, pidx, out, n_cams, n_obs);
}